// GCN_30021821399139
// MI455X (gfx1250) — compile-verified
//
#include <hip/hip_runtime.h>

typedef __attribute__((ext_vector_type(2))) float v2f;
typedef __attribute__((ext_vector_type(8))) float v8f;

#define BN_EPS 1e-5f
#define XS_STRIDE 128

__global__ void k_zero(float* __restrict__ p, int n) {
  int i = blockIdx.x * blockDim.x + threadIdx.x;
  if (i < n) p[i] = 0.0f;
}

__global__ void k_deg_count(const int* __restrict__ src, const int* __restrict__ dst,
                            float* __restrict__ doutc, float* __restrict__ dinc, int E) {
  int e = blockIdx.x * blockDim.x + threadIdx.x;
  if (e < E) {
    atomicAdd(&doutc[src[e]], 1.0f);
    atomicAdd(&dinc[dst[e]], 1.0f);
  }
}

__global__ void k_deg_norm(const float* __restrict__ c, float* __restrict__ o, int N) {
  int i = blockIdx.x * blockDim.x + threadIdx.x;
  if (i < N) o[i] = rsqrtf(fmaxf(c[i], 1.0f));
}

// Xs[n][k] = X[n][k] * dnorm[n], zero-padded to Npad rows / Kpad cols (stride XS_STRIDE).
__global__ void k_prescale(const float* __restrict__ X, const float* __restrict__ dnorm,
                           float* __restrict__ Xs, int N, int Fi, int Kpad, int Npad) {
  int idx = blockIdx.x * blockDim.x + threadIdx.x;
  int total = Npad * Kpad;
  if (idx >= total) return;
  int n = idx / Kpad;
  int k = idx - n * Kpad;
  float v = (n < N && k < Fi) ? X[(long)n * Fi + k] * dnorm[n] : 0.0f;
  Xs[(long)n * XS_STRIDE + k] = v;
}

// Wp[Kpad x Fopad] = zero-padded W[Fi x Fo].
__global__ void k_padW(const float* __restrict__ W, float* __restrict__ Wp,
                       int Fi, int Fo, int Kpad, int Fopad) {
  int idx = blockIdx.x * blockDim.x + threadIdx.x;
  int total = Kpad * Fopad;
  if (idx >= total) return;
  int k = idx / Fopad;
  int c = idx - k * Fopad;
  Wp[idx] = (k < Fi && c < Fo) ? W[(long)k * Fo + c] : 0.0f;
}

// Y[Npad x Fopad] = Xs[Npad x Kpad] @ Wp[Kpad x Fopad].
// One wave32 owns a 16x16 tile via V_WMMA_F32_16X16X4_F32; all loads/stores unconditional.
__global__ void k_gemm_wmma(const float* __restrict__ Xs, const float* __restrict__ Wp,
                            float* __restrict__ Y, int Kpad, int Fopad, int tilesN,
                            int nwaves) {
  int wave = (int)((blockIdx.x * (unsigned)blockDim.x + threadIdx.x) >> 5);
  if (wave >= nwaves) return;            // wave-uniform: EXEC all-ones at WMMA
  int lane = threadIdx.x & 31;
  int tm = wave / tilesN;
  int tn = wave - tm * tilesN;
  int half = lane >> 4;                  // 0: lanes 0-15, 1: lanes 16-31
  int l16  = lane & 15;
  int arow = tm * 16 + l16;
  int bcol = tn * 16 + l16;
  // A lane layout: VGPR0 = K=k0+2*half, VGPR1 = K=k0+2*half+1 (contiguous -> b64 load)
  const float* ap = Xs + (long)arow * XS_STRIDE + 2 * half;
  // B lane layout mirrors A across rows of W
  const float* bp = Wp + (long)(2 * half) * Fopad + bcol;
  v8f acc = {};
  for (int k0 = 0; k0 < Kpad; k0 += 4) {
    v2f av = *(const v2f*)(ap + k0);
    v2f bv;
    bv.x = bp[(long)k0 * Fopad];
    bv.y = bp[(long)k0 * Fopad + Fopad];
    acc = __builtin_amdgcn_wmma_f32_16x16x4_f32(false, av, false, bv, (short)0, acc,
                                                false, false);
  }
  union { v8f v; float f[8]; } u; u.v = acc;
  // D layout: VGPR v -> row = tm*16 + v + 8*half
  float* yp = Y + (long)(tm * 16 + 8 * half) * Fopad + bcol;
#pragma unroll
  for (int v = 0; v < 8; ++v) yp[(long)v * Fopad] = u.f[v];
}

// agg[dst[e], f] += XW[src[e], f]; XW has padded stride Fopad.
__global__ void k_scatter(const int* __restrict__ src, const int* __restrict__ dst,
                          const float* __restrict__ XW, float* __restrict__ agg,
                          int E, int Fo, int Fopad) {
  int idx = blockIdx.x * blockDim.x + threadIdx.x;
  int total = E * Fo;
  if (idx >= total) return;
  int e = idx / Fo;
  int f = idx - e * Fo;
  atomicAdd(&agg[(long)dst[e] * Fo + f], XW[(long)src[e] * Fopad + f]);
}

// h = agg*din + b (in place); per-feature sum / sumsq via LDS block partials.
__global__ void k_finalize_stats(float* __restrict__ agg, const float* __restrict__ dinn,
                                 const float* __restrict__ b, float* __restrict__ sum,
                                 float* __restrict__ sumsq, int N, int Fo) {
  extern __shared__ float ls[];          // 2*Fo floats
  for (int i = threadIdx.x; i < 2 * Fo; i += blockDim.x) ls[i] = 0.0f;
  __syncthreads();
  int idx = blockIdx.x * blockDim.x + threadIdx.x;
  if (idx < N * Fo) {
    int n = idx / Fo;
    int f = idx - n * Fo;
    float v = agg[idx] * dinn[n] + b[f];
    agg[idx] = v;
    atomicAdd(&ls[f], v);
    atomicAdd(&ls[Fo + f], v * v);
  }
  __syncthreads();
  for (int i = threadIdx.x; i < 2 * Fo; i += blockDim.x) {
    float p = ls[i];
    if (i < Fo) atomicAdd(&sum[i], p);
    else        atomicAdd(&sumsq[i - Fo], p);
  }
}

__global__ void k_bn_affine(const float* __restrict__ sum, const float* __restrict__ sumsq,
                            const float* __restrict__ g, const float* __restrict__ be,
                            float* __restrict__ scale, float* __restrict__ shift,
                            int N, int Fo) {
  int f = blockIdx.x * blockDim.x + threadIdx.x;
  if (f >= Fo) return;
  float inv = 1.0f / (float)N;
  float mu = sum[f] * inv;
  float var = sumsq[f] * inv - mu * mu;
  float s = g[f] * rsqrtf(var + BN_EPS);
  scale[f] = s;
  shift[f] = be[f] - mu * s;
}

__global__ void k_apply_prelu(const float* __restrict__ h2, const float* __restrict__ scale,
                              const float* __restrict__ shift, const float* __restrict__ aptr,
                              float* __restrict__ hout, int N, int Fo) {
  int idx = blockIdx.x * blockDim.x + threadIdx.x;
  if (idx >= N * Fo) return;
  int f = idx % Fo;
  float v = h2[idx] * scale[f] + shift[f];
  float a = aptr[0];
  hout[idx] = (v >= 0.0f) ? v : a * v;
}

// Column sums with LDS block partials (Fo <= 8 here).
__global__ void k_colsum(const float* __restrict__ h, float* __restrict__ csum, int N, int Fo) {
  __shared__ float ls[8];
  if (threadIdx.x < (unsigned)Fo) ls[threadIdx.x] = 0.0f;
  __syncthreads();
  int n = blockIdx.x * blockDim.x + threadIdx.x;
  if (n < N) {
    for (int f = 0; f < Fo; ++f) atomicAdd(&ls[f], h[(long)n * Fo + f]);
  }
  __syncthreads();
  if (threadIdx.x < (unsigned)Fo) atomicAdd(&csum[threadIdx.x], ls[threadIdx.x]);
}

__global__ void k_head(const float* __restrict__ csum, const float* __restrict__ Wc,
                       const float* __restrict__ bc, float* __restrict__ out, int N, int F) {
  if (blockIdx.x == 0 && threadIdx.x < 2) {
    int c = threadIdx.x;
    float inv = 1.0f / (float)N;
    float acc = bc[c];
    for (int f = 0; f < F; ++f) acc += (csum[f] * inv) * Wc[f * 2 + c];
    out[c] = acc;
  }
}

extern "C" void kernel_launch(void* const* d_in, const int* in_sizes, int n_in,
                              void* d_out, int out_size, void* d_ws, size_t ws_size,
                              hipStream_t stream) {
  const float* feat = (const float*)d_in[0];
  const int*   src  = (const int*)d_in[1];
  const int*   dst  = (const int*)d_in[2];
  const float *W[6], *bb[6], *gg[6], *be[6];
  for (int i = 0; i < 6; ++i) {
    W[i]  = (const float*)d_in[3 + 4 * i];
    bb[i] = (const float*)d_in[4 + 4 * i];
    gg[i] = (const float*)d_in[5 + 4 * i];
    be[i] = (const float*)d_in[6 + 4 * i];
  }
  const float* aptr = (const float*)d_in[27];
  const float* Wc   = (const float*)d_in[28];
  const float* bc   = (const float*)d_in[29];

  const int N = in_sizes[0] / 128;
  const int E = in_sizes[1];
  static const int FiA[6] = {128, 96, 48, 24, 12, 6};
  static const int FoA[6] = {96, 48, 24, 12, 6, 3};

  const int T = 256;
  auto cdiv = [](long a, long b) { return (int)((a + b - 1) / b); };
  const int tilesM = cdiv(N, 16);
  const int Npad = tilesM * 16;

  float* ws    = (float*)d_ws;
  float* doutc = ws;                          // N
  float* dinc  = doutc + N;                   // N
  float* doutn = dinc + N;                    // N
  float* dinn  = doutn + N;                   // N
  float* Xs    = dinn + N;                    // Npad * 128 : pre-scaled, padded X
  float* Wp    = Xs + (size_t)Npad * 128;     // 128*96     : padded weights
  float* bufA  = Wp + 128 * 96;               // Npad * 96  : GEMM output (Fopad stride)
  float* bufB  = bufA + (size_t)Npad * 96;    // N * 96     : agg / h2 (in-place)
  float* bufH  = bufB + (size_t)N * 96;       // N * 96     : post-BN/PReLU features
  float* sum   = bufH + (size_t)N * 96;       // 96
  float* sumsq = sum + 96;                    // 96
  float* scale = sumsq + 96;                  // 96
  float* shift = scale + 96;                  // 96
  float* csum  = shift + 96;                  // 96 (only 3 used)

  // Degrees (re-zeroed every call: deterministic, no cross-call state).
  k_zero<<<cdiv(2L * N, T), T, 0, stream>>>(doutc, 2 * N);
  k_deg_count<<<cdiv(E, T), T, 0, stream>>>(src, dst, doutc, dinc, E);
  k_deg_norm<<<cdiv(N, T), T, 0, stream>>>(doutc, doutn, N);
  k_deg_norm<<<cdiv(N, T), T, 0, stream>>>(dinc, dinn, N);

  const float* hin = feat;
  for (int L = 0; L < 6; ++L) {
    int Fi = FiA[L], Fo = FoA[L];
    int Kpad = (Fi + 3) & ~3;
    int tilesN = cdiv(Fo, 16);
    int Fopad = tilesN * 16;
    int nwaves = tilesM * tilesN;

    k_prescale<<<cdiv((long)Npad * Kpad, T), T, 0, stream>>>(hin, doutn, Xs, N, Fi, Kpad, Npad);
    k_padW<<<cdiv((long)Kpad * Fopad, T), T, 0, stream>>>(W[L], Wp, Fi, Fo, Kpad, Fopad);
    k_gemm_wmma<<<cdiv((long)nwaves * 32, T), T, 0, stream>>>(Xs, Wp, bufA, Kpad, Fopad,
                                                              tilesN, nwaves);
    k_zero<<<cdiv((long)N * Fo, T), T, 0, stream>>>(bufB, N * Fo);
    k_zero<<<1, T, 0, stream>>>(sum, 192);     // sum + sumsq (contiguous)
    k_scatter<<<cdiv((long)E * Fo, T), T, 0, stream>>>(src, dst, bufA, bufB, E, Fo, Fopad);
    k_finalize_stats<<<cdiv((long)N * Fo, T), T, 2 * Fo * sizeof(float), stream>>>(
        bufB, dinn, bb[L], sum, sumsq, N, Fo);
    k_bn_affine<<<1, 128, 0, stream>>>(sum, sumsq, gg[L], be[L], scale, shift, N, Fo);
    k_apply_prelu<<<cdiv((long)N * Fo, T), T, 0, stream>>>(bufB, scale, shift, aptr, bufH, N, Fo);
    hin = bufH;
  }

  k_zero<<<1, 64, 0, stream>>>(csum, 64);
  k_colsum<<<cdiv(N, T), T, 0, stream>>>(bufH, csum, N, 3);
  k_head<<<1, 32, 0, stream>>>(csum, Wc, bc, (float*)d_out, N, 3);
}